// LocalAttention2d_64974265254120
// MI455X (gfx1250) — compile-verified
//
#include <hip/hip_runtime.h>
#include <math.h>

typedef float v2f __attribute__((ext_vector_type(2)));
typedef float v8f __attribute__((ext_vector_type(8)));

#define B_   8
#define T_   256
#define D_   256     // q channels
#define CSZ  512     // c_size
#define HW_  128
#define PIX  (HW_ * HW_)          // 16384
#define NTOK (B_ * T_)            // 2048
#define RWIN 9
#define KWIN 81

// ---------------------------------------------------------------------------
// Tiled transpose: q (B, D, H*W) -> qT (B, H*W, D). Makes the window gather
// coalesced along D (which is the hot axis of the attention inner loops).
// ---------------------------------------------------------------------------
__global__ __launch_bounds__(256)
void transpose_kernel(const float* __restrict__ q, float* __restrict__ qT) {
    __shared__ float tile[32][33];
    const int b    = blockIdx.z;
    const int pix0 = blockIdx.x * 32;   // 512 tiles
    const int d0   = blockIdx.y * 32;   // 8 tiles
    const long base = (long)b * (long)(D_ * PIX);
    const int tx = threadIdx.x;         // 0..31
    const int ty = threadIdx.y;         // 0..7
    #pragma unroll
    for (int i = 0; i < 32; i += 8)
        tile[ty + i][tx] = q[base + (long)(d0 + ty + i) * PIX + pix0 + tx];
    __syncthreads();
    #pragma unroll
    for (int i = 0; i < 32; i += 8)
        qT[base + (long)(pix0 + ty + i) * D_ + d0 + tx] = tile[tx][ty + i];
}

// ---------------------------------------------------------------------------
// Fused fp32 WMMA GEMM:
//   u[n,d] = sum_c ct[n,c] * Wa[c,d]           (n_tile 0..15)
//   h[n,p] = tanh( sum_c ct[n,c] * Wp[p,c] )   (n_tile 16..31)
// One 16x16 tile per wave, K stepped by 4 with V_WMMA_F32_16X16X4_F32.
// Layouts per ISA 7.12.2:
//   A: M = lane%16, K = vgpr + 2*(lane/16)
//   B: K = vgpr + 2*(lane/16), N = lane%16
//   C: N = lane%16, M = vgpr + 8*(lane/16)
// W_a-vs-W_p selection hoisted pre-loop as (base, step, elem-offset); the
// K-loop is software-pipelined (1-stage double buffer) so iteration i+1's
// loads are in flight while WMMA(i) executes (avoids s_wait_loadcnt 0x0
// directly in front of every WMMA).
// ---------------------------------------------------------------------------
__global__ __launch_bounds__(256)
void gemm_kernel(const float* __restrict__ ct,   // (NTOK, 512)
                 const float* __restrict__ Wa,   // (512, 256)
                 const float* __restrict__ Wp,   // (256, 512)
                 float* __restrict__ u,          // (NTOK, 256)
                 float* __restrict__ hT)         // (NTOK, 256), tanh applied
{
    const int lane  = threadIdx.x & 31;
    const int wave  = threadIdx.x >> 5;
    const int m0    = blockIdx.x * 16;           // 128 M-tiles
    const int ntile = blockIdx.y * 8 + wave;     // 0..31
    const bool isU  = (ntile < 16);
    const int n0    = (isU ? ntile : (ntile - 16)) * 16;

    const int mrow  = lane & 15;                 // M (for A) and N (for B/C)
    const int khalf = (lane >> 4) * 2;           // K half-select: 0 or 2

    const float* arow = ct + (long)(m0 + mrow) * CSZ + khalf;

    // Uniform B addressing: Bv[v] = B[k0 + khalf + v][n0 + mrow]
    //  isU : B[k][n] = Wa[k*256 + n]        -> step 4*256, elem offset 256
    //  !isU: B[k][p] = Wp[p*512 + k]        -> step 4,     elem offset 1
    const float* bp;
    long bstep;
    int  b2off;
    if (isU) {
        bp    = Wa + (long)khalf * D_ + n0 + mrow;
        bstep = 4L * D_;
        b2off = D_;
    } else {
        bp    = Wp + (long)(n0 + mrow) * CSZ + khalf;
        bstep = 4L;
        b2off = 1;
    }

    // ---- prologue: stage 0 loads ----
    v2f A_c, B_c;
    A_c[0] = arow[0];
    A_c[1] = arow[1];
    B_c[0] = bp[0];
    B_c[1] = bp[b2off];
    bp += bstep;

    v8f acc = {};
    #pragma unroll 8
    for (int k0 = 4; k0 < CSZ; k0 += 4) {
        // issue next iteration's loads before consuming current registers
        v2f A_n, B_n;
        A_n[0] = arow[k0];
        A_n[1] = arow[k0 + 1];
        B_n[0] = bp[0];
        B_n[1] = bp[b2off];
        bp += bstep;
        acc = __builtin_amdgcn_wmma_f32_16x16x4_f32(
            false, A_c, false, B_c, (short)0, acc, false, false);
        A_c = A_n;
        B_c = B_n;
    }
    acc = __builtin_amdgcn_wmma_f32_16x16x4_f32(
        false, A_c, false, B_c, (short)0, acc, false, false);

    float* dst = isU ? u : hT;
    const int mhi = (lane >> 4) * 8;
    #pragma unroll
    for (int v = 0; v < 8; ++v) {
        const int m = m0 + v + mhi;
        float val = acc[v];
        if (!isU) val = tanhf(val);
        dst[(long)m * D_ + n0 + mrow] = val;
    }
}

// ---------------------------------------------------------------------------
// Per-token local attention. One block (256 threads = 8 wave32s) per token.
// ---------------------------------------------------------------------------
__device__ __forceinline__ float waveReduceSum(float v) {
    #pragma unroll
    for (int o = 16; o > 0; o >>= 1) v += __shfl_xor(v, o, 32);
    return v;
}

__global__ __launch_bounds__(256)
void attn_kernel(const float* __restrict__ src,  // qT (pix-major) or q (d-major)
                 const float* __restrict__ u,    // (NTOK, 256)
                 const float* __restrict__ h,    // (NTOK, 256) = tanh(ct@Wp^T)
                 const float* __restrict__ Vp,   // (2, 256)
                 float* __restrict__ out,        // (NTOK, 256)
                 int stride_pix, int stride_d)
{
    const int token = blockIdx.x;       // 0..2047
    const int b     = token >> 8;       // T_ = 256
    const int tid   = threadIdx.x;      // 0..255
    const int lane  = tid & 31;
    const int wave  = tid >> 5;
    const long base = (long)b * (long)(D_ * PIX);

    __shared__ float u_s[D_];
    __shared__ float red[16];
    __shared__ float pt_s[2];
    __shared__ int   rr_s[RWIN], cc_s[RWIN];
    __shared__ float gr_s[RWIN], gc_s[RWIN];
    __shared__ int   pix_s[KWIN];
    __shared__ float g_s[KWIN];
    __shared__ float a_s[KWIN];
    __shared__ float w_s[KWIN];

    // ---- load u row; predictive alignment p_t via block reduction ----
    const float hv = h[(long)token * D_ + tid];
    u_s[tid] = u[(long)token * D_ + tid];
    float s0 = hv * Vp[tid];
    float s1 = hv * Vp[D_ + tid];
    s0 = waveReduceSum(s0);
    s1 = waveReduceSum(s1);
    if (lane == 0) { red[wave] = s0; red[8 + wave] = s1; }
    __syncthreads();
    if (tid == 0) {
        float t0 = 0.f, t1 = 0.f;
        #pragma unroll
        for (int i = 0; i < 8; ++i) { t0 += red[i]; t1 += red[8 + i]; }
        pt_s[0] = 128.f / (1.f + expf(-t0));
        pt_s[1] = 128.f / (1.f + expf(-t1));
    }
    __syncthreads();
    const float p0 = pt_s[0], p1 = pt_s[1];

    // ---- window rows/cols (+1 shift into NaN-padded frame), Gaussian terms
    if (tid < RWIN) {
        int rr = (int)rintf(p0) + (tid - RWIN / 2) + 1;
        rr = min(max(rr, 0), HW_ + 1) % (HW_ + 1);       // 129 -> 0 (pad row)
        rr_s[tid] = rr;
        const float rf = (float)max(rr - 1, 0);
        const float dr = (rf - p0) * 0.25f;              // / (R//2 == 4)
        gr_s[tid] = -2.f * dr * dr;

        int cc = (int)rintf(p1) + (tid - RWIN / 2) + 1;
        cc = min(max(cc, 0), HW_ + 1) % (HW_ + 1);
        cc_s[tid] = cc;
        const float cf = (float)max(cc - 1, 0);
        const float dc = (cf - p1) * 0.25f;
        gc_s[tid] = -2.f * dc * dc;
    }
    __syncthreads();
    if (tid < KWIN) {
        const int i = tid / RWIN, j = tid % RWIN;
        const int rr = rr_s[i], cc = cc_s[j];
        pix_s[tid] = (rr >= 1 && cc >= 1) ? ((rr - 1) * HW_ + (cc - 1)) : -1;
        g_s[tid]   = expf(gr_s[i] + gc_s[j]);
    }
    __syncthreads();

    // ---- scores a[k] = qg[k,:] . u  (wave-parallel over k) ----
    for (int k = wave; k < KWIN; k += 8) {
        const int pix = pix_s[k];
        float acc = 0.f;
        if (pix >= 0) {
            const float* p = src + base + (long)pix * stride_pix;
            #pragma unroll
            for (int it = 0; it < 8; ++it) {
                const int d = lane + 32 * it;
                acc += p[(long)d * stride_d] * u_s[d];
            }
        }
        acc = waveReduceSum(acc);
        if (lane == 0) a_s[k] = (pix >= 0) ? acc : -__builtin_inff();
    }
    __syncthreads();

    // ---- masked softmax * gaussian (wave 0) ----
    if (wave == 0) {
        float m = -__builtin_inff();
        for (int k = lane; k < KWIN; k += 32) m = fmaxf(m, a_s[k]);
        #pragma unroll
        for (int o = 16; o > 0; o >>= 1) m = fmaxf(m, __shfl_xor(m, o, 32));
        float s = 0.f;
        for (int k = lane; k < KWIN; k += 32)
            if (pix_s[k] >= 0) s += expf(a_s[k] - m);
        s = waveReduceSum(s);
        const float inv = 1.f / s;
        for (int k = lane; k < KWIN; k += 32)
            w_s[k] = (pix_s[k] >= 0) ? expf(a_s[k] - m) * inv * g_s[k] : 0.f;
    }
    __syncthreads();

    // ---- out[d] = sum_k w[k] * qg[k,d]  (thread = d, coalesced over lanes)
    float acc = 0.f;
    for (int k = 0; k < KWIN; ++k) {
        const int pix = pix_s[k];
        if (pix >= 0)
            acc += w_s[k] * src[base + (long)pix * stride_pix + (long)tid * stride_d];
    }
    out[(long)token * D_ + tid] = acc;
}

// ---------------------------------------------------------------------------
extern "C" void kernel_launch(void* const* d_in, const int* in_sizes, int n_in,
                              void* d_out, int out_size, void* d_ws, size_t ws_size,
                              hipStream_t stream) {
    const float* q  = (const float*)d_in[0];   // (8, 256, 128, 128)
    const float* ct = (const float*)d_in[1];   // (8, 256, 512)
    const float* Wa = (const float*)d_in[2];   // (512, 256)
    const float* Wp = (const float*)d_in[3];   // (256, 512)
    const float* Vp = (const float*)d_in[4];   // (2, 256)
    float* out = (float*)d_out;                // (8, 256, 256)

    const size_t qElems = (size_t)B_ * D_ * PIX;   // 33,554,432
    const size_t uElems = (size_t)NTOK * D_;       // 524,288
    float* ws = (float*)d_ws;

    float* u;
    float* hT;
    const float* src;
    int stride_pix, stride_d;

    const size_t needT = (qElems + 2 * uElems) * sizeof(float);
    if (ws_size >= needT) {
        // transposed path: gather fully coalesced along D
        float* qT = ws;
        u  = ws + qElems;
        hT = ws + qElems + uElems;
        transpose_kernel<<<dim3(PIX / 32, D_ / 32, B_), dim3(32, 8), 0, stream>>>(q, qT);
        src = qT; stride_pix = D_; stride_d = 1;
    } else {
        // fallback: strided gather straight from q
        u  = ws;
        hT = ws + uElems;
        src = q; stride_pix = 1; stride_d = PIX;
    }

    gemm_kernel<<<dim3(NTOK / 16, 4), 256, 0, stream>>>(ct, Wa, Wp, u, hT);
    attn_kernel<<<NTOK, 256, 0, stream>>>(src, u, hT, Vp, out, stride_pix, stride_d);
}